// AutomatonNetwork_67156108640980
// MI455X (gfx1250) — compile-verified
//
#include <hip/hip_runtime.h>
#include <hip/hip_bf16.h>

typedef float v2f __attribute__((ext_vector_type(2)));
typedef float v8f __attribute__((ext_vector_type(8)));

constexpr int SDIM  = 512;            // automaton states
constexpr int TLEN  = 4096;           // tokens
constexpr int GBLK  = 64;             // token blocks
constexpr int LBLK  = TLEN / GBLK;    // 64 tokens per block
constexpr int SP    = 528;            // augmented + padded (33 tiles of 16)
constexpr int KT    = SP / 16;        // 33 k-tiles
constexpr int KS    = SP / 4;         // 132 K=4 sub-steps
constexpr int ROWS  = 64;             // product rows per workgroup (slab)
constexpr int ITILES= ROWS / 16;      // 4
constexpr int JPW   = 9;              // j-tiles per wave (overlapping groups of 8)

constexpr int A_ELEMS = ITILES * KS * 32 * 2;   // 33792 floats (135 KB)
constexpr int B_ELEMS = 4 * KT * 32 * 2;        //  8448 floats ( 33 KB)
constexpr size_t SMEM_BYTES = (size_t)(A_ELEMS + B_ELEMS) * sizeof(float);

__device__ __forceinline__ int a_addr(int i, int ks, int lane) {
    return ((i * KS + ks) * 32 + lane) * 2;
}
__device__ __forceinline__ int b_addr(int ks4, int j, int lane) {
    return ((ks4 * KT + j) * 32 + lane) * 2;
}

// ---------------------------------------------------------------------------
// Phase 1: each workgroup evolves a 64-row slab of its token-block's running
// augmented product P <- P * A_t via v_wmma_f32_16x16x4_f32, P slab in LDS.
// ---------------------------------------------------------------------------
__global__ __launch_bounds__(512)
void automaton_phase1(const int* __restrict__ tokens,
                      const float* __restrict__ tmats,   // [V][512][512]
                      const float* __restrict__ pvecs,   // [V][512]
                      float* __restrict__ mseg)          // [G][512][528]
{
    extern __shared__ float smem[];
    float* As = smem;                 // A slab, fragment layout
    float* Bs = smem + A_ELEMS;       // staged B k-slab, fragment layout

    const int tid  = threadIdx.x;
    const int lane = tid & 31;
    const int wave = tid >> 5;        // 0..15
    const int iT   = wave & 3;        // i-tile within slab
    const int jg   = wave >> 2;       // j group (tiles jg*8 .. jg*8+8)
    const int blk  = blockIdx.y;      // token block
    const int row0 = blockIdx.x * ROWS;
    const int t0   = blk * LBLK;

    // ---- initialize A slab with augmented A_{t0} (rows row0..row0+63 < 512)
    {
        const int tok = tokens[t0];
        const float* M = tmats + (size_t)tok * SDIM * SDIM;
        const float* q = pvecs + (size_t)tok * SDIM;
        for (int e = tid; e < ROWS * SP; e += 512) {
            int r = e / SP;
            int c = e - r * SP;
            int grow = row0 + r;
            float v;
            if (c < SDIM)       v = M[(size_t)grow * SDIM + c];
            else if (c == SDIM) v = q[grow];
            else                v = 0.0f;
            int cc = c & 3;
            int ln = (r & 15) | ((cc >> 1) << 4);
            As[a_addr(r >> 4, c >> 2, ln) + (cc & 1)] = v;
        }
    }
    __syncthreads();

    for (int s = 1; s < LBLK; ++s) {
        const int tok = tokens[t0 + s];
        const float* M = tmats + (size_t)tok * SDIM * SDIM;
        const float* q = pvecs + (size_t)tok * SDIM;

        v8f acc[JPW] = {};

        for (int kt = 0; kt < KT; ++kt) {
            __syncthreads();
            // stage augmented B rows [kt*16, kt*16+16) x cols [0,528)
            for (int e = tid; e < 16 * SP; e += 512) {
                int kl = e / SP;
                int c  = e - kl * SP;
                int k  = kt * 16 + kl;
                float v;
                if (k < SDIM) {
                    v = (c < SDIM) ? M[(size_t)k * SDIM + c]
                                   : ((c == SDIM) ? q[k] : 0.0f);
                } else {
                    v = (k == SDIM && c == SDIM) ? 1.0f : 0.0f;
                }
                int kk = k & 3;
                int ln = (c & 15) | ((kk & 1) << 4);
                Bs[b_addr((k >> 2) & 3, c >> 4, ln) + (kk >> 1)] = v;
            }
            __syncthreads();

            #pragma unroll
            for (int ks4 = 0; ks4 < 4; ++ks4) {
                const int ks = kt * 4 + ks4;
                v2f a = *(const v2f*)&As[a_addr(iT, ks, lane)];
                #pragma unroll
                for (int j = 0; j < JPW; ++j) {
                    const int jt = jg * 8 + j;      // <= 32, compile-time safe
                    v2f b = *(const v2f*)&Bs[b_addr(ks4, jt, lane)];
                    acc[j] = __builtin_amdgcn_wmma_f32_16x16x4_f32(
                        false, a, false, b, (short)0, acc[j], false, false);
                }
            }
        }
        __syncthreads();   // all A-slab reads complete before overwrite

        if (s == LBLK - 1) {
            // final segment product -> global, plain row-major [512][528]
            float* out = mseg + (size_t)blk * SDIM * SP;
            #pragma unroll
            for (int j = 0; j < JPW; ++j) {
                const int jt = jg * 8 + j;
                #pragma unroll
                for (int r = 0; r < 8; ++r) {
                    int m = iT * 16 + r + ((lane >> 4) << 3);
                    int c = jt * 16 + (lane & 15);
                    out[(size_t)(row0 + m) * SP + c] = acc[j][r];
                }
            }
        } else {
            // scatter D tiles back into A-fragment layout for next step
            #pragma unroll
            for (int j = 0; j < JPW; ++j) {
                const int jt = jg * 8 + j;
                #pragma unroll
                for (int r = 0; r < 8; ++r) {
                    int m  = iT * 16 + r + ((lane >> 4) << 3);
                    int c  = jt * 16 + (lane & 15);
                    int cc = c & 3;
                    int ln = (m & 15) | ((cc >> 1) << 4);
                    As[a_addr(m >> 4, c >> 2, ln) + (cc & 1)] = acc[j][r];
                }
            }
        }
        __syncthreads();
    }
}

// ---------------------------------------------------------------------------
// Phase 2: fold [start_vector, start_prob] through the 64 segment matrices.
// Augmented row 512 (== e_512) handled analytically.
// ---------------------------------------------------------------------------
__global__ __launch_bounds__(512)
void automaton_phase2(const float* __restrict__ mseg,
                      const float* __restrict__ start_prob,
                      const float* __restrict__ start_vec,
                      const float* __restrict__ finals,
                      float* __restrict__ out)
{
    __shared__ float w[SP];
    __shared__ float wn[SP];
    __shared__ float red[512];
    const int tid = threadIdx.x;

    w[tid] = start_vec[tid];
    if (tid == 0) w[SDIM] = start_prob[0];
    if (tid < SP - SDIM - 1) w[SDIM + 1 + tid] = 0.0f;
    __syncthreads();

    for (int g = 0; g < GBLK; ++g) {
        const float* Mb = mseg + (size_t)g * SDIM * SP;
        float acc0 = 0.0f, acc1 = 0.0f;
        for (int i = 0; i < SDIM; ++i) {
            float wi = w[i];
            acc0 += wi * Mb[(size_t)i * SP + tid];
            if (tid < SP - SDIM) acc1 += wi * Mb[(size_t)i * SP + SDIM + tid];
        }
        wn[tid] = acc0;
        if (tid < SP - SDIM)
            wn[SDIM + tid] = acc1 + ((tid == 0) ? w[SDIM] : 0.0f);
        __syncthreads();
        w[tid] = wn[tid];
        if (tid < SP - SDIM) w[SDIM + tid] = wn[SDIM + tid];
        __syncthreads();
    }

    red[tid] = w[tid] * finals[tid];
    __syncthreads();
    for (int off = 256; off > 0; off >>= 1) {
        if (tid < off) red[tid] += red[tid + off];
        __syncthreads();
    }
    if (tid == 0) out[0] = 1.0f - expf(w[SDIM] + red[0]);
}

extern "C" void kernel_launch(void* const* d_in, const int* in_sizes, int n_in,
                              void* d_out, int out_size, void* d_ws, size_t ws_size,
                              hipStream_t stream) {
    const int*   tokens     = (const int*)d_in[0];
    const float* start_prob = (const float*)d_in[1];
    const float* start_vec  = (const float*)d_in[2];
    const float* tmats      = (const float*)d_in[3];
    const float* pvecs      = (const float*)d_in[4];
    const float* finals     = (const float*)d_in[5];
    float*       outp       = (float*)d_out;
    float*       mseg       = (float*)d_ws;   // GBLK*512*528 floats ~ 69 MB

    dim3 grid(SDIM / ROWS, GBLK);             // (8, 64) = 512 workgroups
    automaton_phase1<<<grid, 512, SMEM_BYTES, stream>>>(tokens, tmats, pvecs, mseg);
    automaton_phase2<<<1, 512, 0, stream>>>(mseg, start_prob, start_vec, finals, outp);
}